// RQ_38457137168776
// MI455X (gfx1250) — compile-verified
//
#include <hip/hip_runtime.h>
#include <stdint.h>

// Residual VQ (3 codebooks) for MI455X / gfx1250, wave32.
//   out[n]  = q1 + q2 + q3 = x - r_final
//   loss[s] = 1.25 * mean(residual_after_s^2)   (beta=1, gamma=0.25 forward value)
// Nearest codeword via argmax(r.w - 0.5||w||^2) using V_WMMA_F32_16X16X4_F32.
// Codebook tiles are double-buffered in LDS via GLOBAL_LOAD_ASYNC_TO_LDS_B128
// and fenced with S_WAIT_ASYNCCNT (async loads complete in order).

typedef __attribute__((ext_vector_type(2))) float v2f;
typedef __attribute__((ext_vector_type(4))) float f4;
typedef __attribute__((ext_vector_type(8))) float v8f;

typedef int v4i __attribute__((vector_size(16)));
typedef __attribute__((address_space(1))) v4i gl_v4i;   // global int4
typedef __attribute__((address_space(3))) v4i lds_v4i;  // LDS int4

#define RQ_N     131072
#define RQ_D     256
#define RQ_ROWS  16          // rows (M) per wave tile
#define RQ_PAD   260         // LDS row stride in floats (260 % 64 == 4 -> conflict-free WMMA operand loads)
#define RQ_NT    (RQ_N / RQ_ROWS)   // 8192 row tiles

// ---- CDNA5 async copy: global -> LDS, 16B per lane, ASYNCcnt-tracked --------
__device__ __forceinline__ void async_cp16(const float* g, float* l) {
#if __has_builtin(__builtin_amdgcn_global_load_async_to_lds_b128)
    gl_v4i*  gp = (gl_v4i*)(uintptr_t)g;                 // flat addr == global addr
    lds_v4i* lp = (lds_v4i*)(uint32_t)(uintptr_t)l;      // flat addr[31:0] == LDS offset
    __builtin_amdgcn_global_load_async_to_lds_b128(gp, lp, /*offset=*/0, /*cpol=*/0);
#else
    uint32_t lo = (uint32_t)(uintptr_t)l;
    asm volatile("global_load_async_to_lds_b128 %0, %1, off"
                 :: "v"(lo), "v"(g) : "memory");
#endif
}

__device__ __forceinline__ void wait_async_le32() {
#if __has_builtin(__builtin_amdgcn_s_wait_asynccnt)
    __builtin_amdgcn_s_wait_asynccnt(32);
#else
    asm volatile("s_wait_asynccnt 0x20" ::: "memory");
#endif
    asm volatile("" ::: "memory");
}

__device__ __forceinline__ void wait_async_0() {
#if __has_builtin(__builtin_amdgcn_s_wait_asynccnt)
    __builtin_amdgcn_s_wait_asynccnt(0);
#else
    asm volatile("s_wait_asynccnt 0x0" ::: "memory");
#endif
    asm volatile("" ::: "memory");
}

// Issue one 16x256-f32 tile (16 KB) as 32 async b128 copies (all 32 lanes each).
__device__ __forceinline__ void issue_tile(const float* gbase, float* lbase, int lane) {
    #pragma unroll 8
    for (int it = 0; it < 32; ++it) {
        int t = it * 32 + lane;            // 0..1023 float4 chunks
        int c = t >> 6, d = (t & 63) << 2; // row (codeword), dim
        async_cp16(gbase + c * RQ_D + d, lbase + c * RQ_PAD + d);
    }
}

__global__ __launch_bounds__(32)
void rq_vq3_kernel(const float* __restrict__ x,
                   const float* __restrict__ W1,
                   const float* __restrict__ W2,
                   const float* __restrict__ W3,
                   float* __restrict__ out,
                   float* __restrict__ loss_part) {
    __shared__ float resid [RQ_ROWS * RQ_PAD];
    __shared__ float ctileA[RQ_ROWS * RQ_PAD];
    __shared__ float ctileB[RQ_ROWS * RQ_PAD];
    __shared__ int   idxbuf[RQ_ROWS];

    const int  lane    = threadIdx.x;      // single wave32 per block
    const int  l15     = lane & 15;
    const int  half    = lane >> 4;
    const long rowbase = (long)blockIdx.x * RQ_ROWS;

    // residual tile starts as x tile: async-stage it (completion ordered before tiles)
    issue_tile(x + rowbase * RQ_D, resid, lane);

    const float* Ws[3] = {W1, W2, W3};
    const int    Ks[3] = {1024, 512, 256};

    for (int s = 0; s < 3; ++s) {
        const float* __restrict__ W = Ws[s];
        const int K = Ks[s];

        // running per-row argmax: slot j holds row (j + 8*half), this lane's column
        float bestv[8];
        int   besti[8];
        #pragma unroll
        for (int j = 0; j < 8; ++j) { bestv[j] = -3.402823466e38f; besti[j] = 0; }

        issue_tile(W, ctileA, lane);       // prologue: tile 0 in flight
        int cur = 0;

        for (int cb = 0; cb < K; cb += 16) {
            const bool more = (cb + 16) < K;
            float* curbuf = cur ? ctileB : ctileA;   // runtime select, no LDS-ptr array
            float* nxtbuf = cur ? ctileA : ctileB;
            if (more) issue_tile(W + (long)(cb + 16) * RQ_D, nxtbuf, lane);
            // async loads complete in order: <=32 outstanding => everything
            // except the just-issued next tile (incl. x tile) has landed in LDS.
            if (more) wait_async_le32(); else wait_async_0();

            const float* ctile = curbuf;

            // 0.5*||w||^2 for this lane's column (cb + l15): halves split, shfl-combined
            float nrm = 0.f;
            {
                const float* wr = ctile + l15 * RQ_PAD + half * 128;
                #pragma unroll
                for (int q = 0; q < 32; ++q) {
                    f4 v = *(const f4*)(wr + q * 4);
                    nrm += v.x * v.x + v.y * v.y + v.z * v.z + v.w * v.w;
                }
                nrm += __shfl_xor(nrm, 16, 32);
            }

            v8f c;
            #pragma unroll
            for (int j = 0; j < 8; ++j) c[j] = -0.5f * nrm;

            // A: 16x4 f32 (lane<16: M=lane, K=k0,k0+1; lane>=16: M=lane-16, K=k0+2,k0+3)
            // B: 4x16 f32, same per-lane offset pattern with N in place of M.
            const int koff = half * 2;
            const float* arow = resid + l15 * RQ_PAD + koff;
            const float* brow = ctile + l15 * RQ_PAD + koff;
            #pragma unroll 8
            for (int k0 = 0; k0 < RQ_D; k0 += 4) {
                v2f a = *(const v2f*)(arow + k0);
                v2f b = *(const v2f*)(brow + k0);
                c = __builtin_amdgcn_wmma_f32_16x16x4_f32(
                        false, a, false, b, (short)0, c, false, false);
            }

            const int col = cb + l15;
            #pragma unroll
            for (int j = 0; j < 8; ++j)
                if (c[j] > bestv[j]) { bestv[j] = c[j]; besti[j] = col; }

            cur ^= 1;
        }

        // argmax across the 16 columns in each 16-lane half (smaller index wins ties)
        #pragma unroll
        for (int off = 1; off < 16; off <<= 1) {
            #pragma unroll
            for (int j = 0; j < 8; ++j) {
                float ov = __shfl_xor(bestv[j], off, 32);
                int   oi = __shfl_xor(besti[j], off, 32);
                if (ov > bestv[j] || (ov == bestv[j] && oi < besti[j])) {
                    bestv[j] = ov; besti[j] = oi;
                }
            }
        }
        __syncthreads();
        if (l15 == 0) {                    // lane0 -> rows 0..7, lane16 -> rows 8..15
            #pragma unroll
            for (int j = 0; j < 8; ++j) idxbuf[half * 8 + j] = besti[j];
        }
        __syncthreads();

        // gather winning codewords (L2-hot rows): r -= w; accumulate ||r_new||^2
        float ss = 0.f;
        const int d0 = lane * 8;           // 8 dims per lane
        for (int m = 0; m < RQ_ROWS; ++m) {
            const int k = idxbuf[m];
            const float* wrow = W + (long)k * RQ_D + d0;
            f4 w0 = *(const f4*)(wrow);
            f4 w1 = *(const f4*)(wrow + 4);
            float* rp = resid + m * RQ_PAD + d0;
            f4 r0 = *(const f4*)rp - w0;
            f4 r1 = *(const f4*)(rp + 4) - w1;
            *(f4*)rp       = r0;
            *(f4*)(rp + 4) = r1;
            ss += r0.x * r0.x + r0.y * r0.y + r0.z * r0.z + r0.w * r0.w;
            ss += r1.x * r1.x + r1.y * r1.y + r1.z * r1.z + r1.w * r1.w;
        }
        #pragma unroll
        for (int off = 1; off < 32; off <<= 1) ss += __shfl_xor(ss, off, 32);
        if (lane == 0) loss_part[s * RQ_NT + blockIdx.x] = ss;
        __syncthreads();
    }

    // out = x - r_final (x tile re-read is an L2 hit: touched by this block at launch)
    for (int t = lane; t < RQ_ROWS * 64; t += 32) {
        int m = t >> 6, d = (t & 63) << 2;
        f4 xv = *(const f4*)(x + (rowbase + m) * RQ_D + d);
        f4 rv = *(const f4*)(resid + m * RQ_PAD + d);
        *(f4*)(out + (rowbase + m) * RQ_D + d) = xv - rv;
    }
}

// Deterministic loss finalize: fixed-order strided sum + shared tree reduction.
__global__ __launch_bounds__(256)
void rq_loss_reduce_kernel(const float* __restrict__ part, float* __restrict__ out_loss) {
    __shared__ float sm[256];
    const int tid = threadIdx.x;
    for (int s = 0; s < 3; ++s) {
        float acc = 0.f;
        for (int i = tid; i < RQ_NT; i += 256) acc += part[s * RQ_NT + i];
        sm[tid] = acc;
        __syncthreads();
        for (int w = 128; w > 0; w >>= 1) {
            if (tid < w) sm[tid] += sm[tid + w];
            __syncthreads();
        }
        if (tid == 0)
            out_loss[s] = 1.25f * sm[0] / ((float)RQ_N * (float)RQ_D);
        __syncthreads();
    }
}

extern "C" void kernel_launch(void* const* d_in, const int* in_sizes, int n_in,
                              void* d_out, int out_size, void* d_ws, size_t ws_size,
                              hipStream_t stream) {
    const float* x  = (const float*)d_in[0];
    const float* W1 = (const float*)d_in[1];
    const float* W2 = (const float*)d_in[2];
    const float* W3 = (const float*)d_in[3];
    float* out       = (float*)d_out;
    float* out_loss  = out + (size_t)RQ_N * RQ_D;   // 3 floats after the [N,D] output
    float* loss_part = (float*)d_ws;                // 3 * 8192 floats of scratch

    (void)in_sizes; (void)n_in; (void)out_size; (void)ws_size;

    rq_vq3_kernel<<<RQ_NT, 32, 0, stream>>>(x, W1, W2, W3, out, loss_part);
    rq_loss_reduce_kernel<<<1, 256, 0, stream>>>(loss_part, out_loss);
}